// enc_embedding_dynamic_15994458211144
// MI455X (gfx1250) — compile-verified
//
#include <hip/hip_runtime.h>

// ---------------- problem constants (match reference) ----------------
#define D_MODEL   128
#define B_DIM     128
#define S_DIM     4096
#define F_DIM     8
#define FREQ_DIM  4
#define THR       0.2f
#define MINW      10
#define P_MAX     410            // S//MINW + 1
#define P_PAD     416            // round P_MAX up to 16
#define NPT       (P_PAD / 16)   // 26 patch tiles (25 full + 1 guarded)
#define NWORDS    (S_DIM / 32)   // 128 words of candidate bits

typedef __attribute__((ext_vector_type(2))) float v2f;
typedef __attribute__((ext_vector_type(8))) float v8f;

// ---------------- kernel A: pe[s] = pe_table[s,:] . pe_w + pe_b ----------------
__global__ void pe_kernel(const float* __restrict__ pe_table,
                          const float* __restrict__ pe_w,
                          const float* __restrict__ pe_b,
                          float* __restrict__ pe_out) {
    int s = blockIdx.x * blockDim.x + threadIdx.x;
    if (s >= S_DIM) return;
    const float4* t = (const float4*)(pe_table + (size_t)s * D_MODEL);
    const float4* w = (const float4*)pe_w;
    float acc = 0.f;
#pragma unroll
    for (int i = 0; i < D_MODEL / 4; ++i) {
        float4 a = t[i], b = w[i];
        acc += a.x * b.x + a.y * b.y + a.z * b.z + a.w * b.w;
    }
    pe_out[s] = acc + pe_b[0];
}

// ---------------- kernel B: one block per (b,f) row ----------------
__global__ __launch_bounds__(256)
void row_kernel(const float* __restrict__ x,
                const float* __restrict__ x_mark,
                const float* __restrict__ pe,     // [S] precomputed
                const float* __restrict__ te_w,   // [4]
                const float* __restrict__ te_b,   // [1]
                const float* __restrict__ conv_w, // [128,3]
                const float* __restrict__ conv_b, // [128]
                float* __restrict__ out) {        // [B,F,P_MAX,128]
    __shared__ float    s_xd[S_DIM];
    __shared__ float    s_csum[S_DIM + 1];
    __shared__ float    s_part[256];
    __shared__ unsigned s_cand[NWORDS];
    __shared__ int      s_L[P_PAD];
    __shared__ int      s_start[P_PAD];
    __shared__ int      s_nseg;
    __shared__ int      s_M;

    const int row = blockIdx.x;          // row = b*F + f
    const int b   = row / F_DIM;
    const int f   = row % F_DIM;
    const int tid = threadIdx.x;

    // ---- phase 0: xd[s] = x[b,s,f] + pe[s] + (x_mark[b,s,:].te_w + te_b)
    const float tw0 = te_w[0], tw1 = te_w[1], tw2 = te_w[2], tw3 = te_w[3];
    const float tb  = te_b[0];
    const float* xm = x_mark + (size_t)b * S_DIM * FREQ_DIM;
    const float* xr = x + (size_t)b * S_DIM * F_DIM + f;
    for (int s = tid; s < S_DIM; s += 256) {
        float4 m = *(const float4*)(xm + (size_t)s * FREQ_DIM);
        float te = m.x * tw0 + m.y * tw1 + m.z * tw2 + m.w * tw3 + tb;
        s_xd[s] = xr[(size_t)s * F_DIM] + pe[s] + te;
    }
    __syncthreads();

    // ---- phase 1a: per-thread partial sums (16 elems) + candidate bitmask
    {
        int base = tid * 16;
        float run = 0.f;
#pragma unroll
        for (int i = 0; i < 16; ++i) run += s_xd[base + i];
        s_part[tid] = run;
    }
    if (tid < NWORDS) {
        int j0 = tid * 32;
        unsigned bits = 0;
        float prev = s_xd[j0];
#pragma unroll
        for (int i = 0; i < 32; ++i) {
            int j = j0 + i;                      // diff index
            float cur = (j < S_DIM - 1) ? s_xd[j + 1] : prev;
            if (j < S_DIM - 1 && fabsf(cur - prev) > THR) bits |= (1u << i);
            prev = cur;
        }
        s_cand[tid] = bits;
    }
    __syncthreads();

    // ---- phase 1b (single lane): scan partials + jump-scan cuts via ctz
    if (tid == 0) {
        float run = 0.f;
        for (int i = 0; i < 256; ++i) { float t = s_part[i]; s_part[i] = run; run += t; }

        int start = 0, np = 0, maxL = 0;
        for (;;) {
            int jmin = start + (MINW - 1);       // earliest legal cut index
            if (jmin > S_DIM - 2) break;
            int w = jmin >> 5;
            unsigned word = s_cand[w] & (0xFFFFFFFFu << (jmin & 31));
            while (word == 0) {
                if (++w >= NWORDS) break;
                word = s_cand[w];
            }
            if (word == 0) break;
            int j = (w << 5) + __builtin_ctz(word);   // first cut >= jmin
            int len = j - start + 1;
            s_L[np] = len; s_start[np] = start;
            if (len > maxL) maxL = len;
            ++np; start = j + 1;
        }
        int lastL = S_DIM - start;
        s_L[np] = lastL; s_start[np] = start;
        if (lastL > maxL) maxL = lastL;
        ++np;
        for (int p = np; p < P_PAD; ++p) { s_L[p] = 0; s_start[p] = S_DIM; }
        s_nseg = np;
        s_M = maxL;
        s_csum[0] = 0.f;
    }
    __syncthreads();

    // ---- phase 1c: write Csum with per-thread offsets
    {
        int base = tid * 16;
        float run = s_part[tid];
#pragma unroll
        for (int i = 0; i < 16; ++i) { run += s_xd[base + i]; s_csum[base + i + 1] = run; }
    }
    __syncthreads();

    // ---- phase 2: WMMA over 16-patch x 16-channel tiles ----
    const int lane = tid & 31;
    const int wid  = tid >> 5;                 // 8 waves == 8 channel tiles
    const int nseg = s_nseg;
    const int Mm2  = s_M - 2;
    const float rden = 1.0f / (float)((Mm2 > 1) ? Mm2 : 1);

    const int m    = lane & 15;
    const int half = lane >> 4;                // 0: K={0,1}  1: K={2,3}
    const int ks   = half * 2;
    const int k1   = ks + 1;

    // B fragment, bias, output column: loop-invariant (ct == wid)
    const int ct   = wid;
    const int chan = ct * 16 + m;
    const float w0 = conv_w[chan * 3 + 0];
    const float w1 = conv_w[chan * 3 + 1];
    const float w2 = conv_w[chan * 3 + 2];
    v2f bf;
    bf.x = half ? w2 : w0;                     // B[k][n], k=3 row is zero
    bf.y = half ? 0.0f : w1;
    const float bias = conv_b[chan];
    float* ocol = out + (size_t)row * P_MAX * D_MODEL + chan;

#pragma unroll 2
    for (int pt = 0; pt < NPT - 1; ++pt) {     // 25 full tiles: no P_MAX guard
        int p  = pt * 16 + m;
        int Lp = s_L[p];
        int st = s_start[p];
        int hi0 = st + min(Mm2 + ks, Lp), lo0 = st + min(ks, Lp);
        int hi1 = st + min(Mm2 + k1, Lp), lo1 = st + min(k1, Lp);
        v2f a;
        a.x = s_csum[hi0] - s_csum[lo0];
        float a1 = s_csum[hi1] - s_csum[lo1];
        a.y = (k1 < 3) ? a1 : 0.0f;

        v8f acc = {};
#if defined(__gfx1250__)
        acc = __builtin_amdgcn_wmma_f32_16x16x4_f32(
            false, a, false, bf, (short)0, acc, false, false);
#endif
#pragma unroll
        for (int r = 0; r < 8; ++r) {
            int patch = pt * 16 + r + half * 8;     // C/D layout: vgpr r + half -> M
            float v = acc[r] * rden + bias;
            v = (patch < nseg) ? v : 0.0f;
            __builtin_nontemporal_store(v, ocol + (size_t)patch * D_MODEL);
        }
    }

    // ---- tail tile (pt = 25): patches 400..415, guard against P_MAX = 410
    {
        const int pt = NPT - 1;
        int p  = pt * 16 + m;
        int Lp = s_L[p];
        int st = s_start[p];
        int hi0 = st + min(Mm2 + ks, Lp), lo0 = st + min(ks, Lp);
        int hi1 = st + min(Mm2 + k1, Lp), lo1 = st + min(k1, Lp);
        v2f a;
        a.x = s_csum[hi0] - s_csum[lo0];
        float a1 = s_csum[hi1] - s_csum[lo1];
        a.y = (k1 < 3) ? a1 : 0.0f;

        v8f acc = {};
#if defined(__gfx1250__)
        acc = __builtin_amdgcn_wmma_f32_16x16x4_f32(
            false, a, false, bf, (short)0, acc, false, false);
#endif
#pragma unroll
        for (int r = 0; r < 8; ++r) {
            int patch = pt * 16 + r + half * 8;
            float v = acc[r] * rden + bias;
            v = (patch < nseg) ? v : 0.0f;
            if (patch < P_MAX)
                __builtin_nontemporal_store(v, ocol + (size_t)patch * D_MODEL);
        }
    }
}

extern "C" void kernel_launch(void* const* d_in, const int* in_sizes, int n_in,
                              void* d_out, int out_size, void* d_ws, size_t ws_size,
                              hipStream_t stream) {
    const float* x        = (const float*)d_in[0];
    const float* x_mark   = (const float*)d_in[1];
    const float* pe_table = (const float*)d_in[2];
    const float* pe_w     = (const float*)d_in[3];
    const float* pe_b     = (const float*)d_in[4];
    const float* te_w     = (const float*)d_in[5];
    const float* te_b     = (const float*)d_in[6];
    const float* conv_w   = (const float*)d_in[7];
    const float* conv_b   = (const float*)d_in[8];
    float* out = (float*)d_out;
    float* pe  = (float*)d_ws;   // 4096 floats = 16 KB of scratch

    pe_kernel<<<(S_DIM + 255) / 256, 256, 0, stream>>>(pe_table, pe_w, pe_b, pe);
    row_kernel<<<B_DIM * F_DIM, 256, 0, stream>>>(x, x_mark, pe, te_w, te_b,
                                                  conv_w, conv_b, out);
}